// VectorQuantizer_18245021073535
// MI455X (gfx1250) — compile-verified
//
#include <hip/hip_runtime.h>

#define NUM_CODES 512
#define CODE_DIM  64
#define HWSZ      4096       // H*W
#define DHW       262144     // D*H*W
#define NTOT      131072     // B*H*W
#define ROWS_PER_BLOCK 256
#define NBLOCKS   (NTOT / ROWS_PER_BLOCK)   // 512
#define ZQ_ELEMS  8388608    // B*D*H*W
#define IDX_OFF   ZQ_ELEMS
#define LOSS_OFF  (ZQ_ELEMS + NTOT)

typedef __attribute__((ext_vector_type(2))) float v2f;
typedef __attribute__((ext_vector_type(8))) float v8f;

// ---------------------------------------------------------------- norms
__global__ __launch_bounds__(256) void vq_norms(const float* __restrict__ emb,
                                                float* __restrict__ norms) {
    int k = blockIdx.x * 256 + threadIdx.x;
    if (k < NUM_CODES) {
        float s = 0.f;
        #pragma unroll 8
        for (int d = 0; d < CODE_DIM; ++d) {
            float v = emb[k * CODE_DIM + d];
            s += v * v;
        }
        norms[k] = s;
    }
}

// ---------------------------------------------------------------- main
// dynamic LDS layout (floats):
//   [0, 32768)        emb (512 x 64)
//   [32768, 33280)    code norms (512)  -- reused as reduction scratch later
//   [33280, 33536)    per-block winning indices (256 ints)
__global__ __launch_bounds__(256) void vq_main(const float* __restrict__ z_e,
                                               const float* __restrict__ emb,
                                               const float* __restrict__ norms,
                                               float* __restrict__ out,
                                               float* __restrict__ partials) {
    extern __shared__ float smem[];
    float* semb  = smem;                       // 32768 floats
    float* snorm = smem + 32768;               // 512 floats
    int*   sidx  = (int*)(smem + 33280);       // 256 ints

    const int tid = threadIdx.x;

    // Stage codebook (b128 loads / ds_store_b128) + norms into LDS.
    {
        const float4* src = (const float4*)emb;
        float4*       dst = (float4*)semb;
        #pragma unroll 4
        for (int i = 0; i < 32; ++i)
            dst[i * 256 + tid] = src[i * 256 + tid];
        snorm[tid]       = norms[tid];
        snorm[tid + 256] = norms[tid + 256];
    }
    __syncthreads();

    const int lane = tid & 31;
    const int wid  = tid >> 5;
    const int m    = lane & 15;     // row within A tile / column within C tile
    const int half = lane >> 4;     // selects K pair within each k-step
    const int blockRow = blockIdx.x * ROWS_PER_BLOCK;
    const int rowBase  = blockRow + wid * 32;   // 32 rows per wave (2 A tiles)

    // A fragments: rows n0 = rowBase + m and n0 + 16; feature stride = H*W.
    const int nrow = rowBase + m;
    const int bb   = nrow >> 12;
    const int rr   = nrow & 4095;
    const float* xb0 = z_e + (size_t)bb * DHW + rr;        // tile 0
    const float* xb1 = xb0 + 16;                           // tile 1 (same b-block)

    v2f a0[16], a1[16];
    #pragma unroll
    for (int j = 0; j < 16; ++j) {
        const int d = 4 * j + 2 * half;
        a0[j].x = xb0[(size_t)d * HWSZ];
        a0[j].y = xb0[(size_t)(d + 1) * HWSZ];
        a1[j].x = xb1[(size_t)d * HWSZ];
        a1[j].y = xb1[(size_t)(d + 1) * HWSZ];
    }

    float best0[8], best1[8];
    int   bidx0[8], bidx1[8];
    #pragma unroll
    for (int r8 = 0; r8 < 8; ++r8) {
        best0[r8] = 3.4e38f; bidx0[r8] = 0;
        best1[r8] = 3.4e38f; bidx1[r8] = 0;
    }

    // 32 tiles of 16 codes; D = 64 -> 2 interleaved chains of 16 f32 WMMAs.
    for (int t = 0; t < 32; ++t) {
        const int code = t * 16 + m;
        const float* bp = &semb[code * CODE_DIM + 2 * half];
        v8f acc0 = {};
        v8f acc1 = {};
        #pragma unroll
        for (int j = 0; j < 16; ++j) {
            v2f bv = *(const v2f*)(bp + 4 * j);
            acc0 = __builtin_amdgcn_wmma_f32_16x16x4_f32(
                false, a0[j], false, bv, (short)0, acc0, false, false);
            acc1 = __builtin_amdgcn_wmma_f32_16x16x4_f32(
                false, a1[j], false, bv, (short)0, acc1, false, false);
        }
        const float cn = snorm[code];
        #pragma unroll
        for (int r8 = 0; r8 < 8; ++r8) {
            float s0 = cn - 2.0f * acc0[r8];
            float s1 = cn - 2.0f * acc1[r8];
            if (s0 < best0[r8]) { best0[r8] = s0; bidx0[r8] = code; }
            if (s1 < best1[r8]) { best1[r8] = s1; bidx1[r8] = code; }
        }
    }

    // Argmin reduction across the 16 lanes holding each output row
    // (xor masks < 16 stay within each half-wave). Tie-break to lower index.
    #pragma unroll
    for (int r8 = 0; r8 < 8; ++r8) {
        float s0 = best0[r8]; int i0 = bidx0[r8];
        float s1 = best1[r8]; int i1 = bidx1[r8];
        #pragma unroll
        for (int off = 1; off < 16; off <<= 1) {
            float os0 = __shfl_xor(s0, off, 32);
            int   oi0 = __shfl_xor(i0, off, 32);
            float os1 = __shfl_xor(s1, off, 32);
            int   oi1 = __shfl_xor(i1, off, 32);
            if (os0 < s0 || (os0 == s0 && oi0 < i0)) { s0 = os0; i0 = oi0; }
            if (os1 < s1 || (os1 == s1 && oi1 < i1)) { s1 = os1; i1 = oi1; }
        }
        if (m == 0) {
            sidx[wid * 32 + r8 + 8 * half]      = i0;   // tile 0 rows
            sidx[wid * 32 + 16 + r8 + 8 * half] = i1;   // tile 1 rows
        }
    }
    __syncthreads();

    // indices output (as float, concatenated after z_q)
    out[IDX_OFF + blockRow + tid] = (float)sidx[tid];

    // z_q gather + loss partial. 256 rows x 64 features = 16384 elems,
    // 64 iters x 256 threads, contiguous 1KB store runs per d.
    const int b0 = blockRow >> 12;
    const int r0 = blockRow & 4095;
    float lsum = 0.f;
    for (int it = 0; it < 64; ++it) {
        const int linear = it * 256 + tid;
        const int d  = linear >> 8;
        const int mm = linear & 255;
        const size_t off = (size_t)b0 * DHW + (size_t)d * HWSZ + (size_t)(r0 + mm);
        const float q  = semb[sidx[mm] * CODE_DIM + d];
        const float x  = z_e[off];
        out[off] = q;                 // z_q_st == z_q numerically
        const float df = q - x;
        lsum += df * df;
    }

    // Block-level deterministic reduction (reuse norm region as scratch).
    float* red = smem + 32768;
    red[tid] = lsum;
    __syncthreads();
    for (int s = 128; s > 0; s >>= 1) {
        if (tid < s) red[tid] += red[tid + s];
        __syncthreads();
    }
    if (tid == 0) partials[blockIdx.x] = red[0];
}

// ---------------------------------------------------------------- finalize
__global__ __launch_bounds__(256) void vq_finalize(const float* __restrict__ partials,
                                                   float* __restrict__ out) {
    __shared__ float red[256];
    const int tid = threadIdx.x;
    red[tid] = partials[tid] + partials[tid + 256];
    __syncthreads();
    for (int s = 128; s > 0; s >>= 1) {
        if (tid < s) red[tid] += red[tid + s];
        __syncthreads();
    }
    if (tid == 0)
        out[LOSS_OFF] = 1.25f * red[0] / (float)ZQ_ELEMS;
}

// ---------------------------------------------------------------- launch
extern "C" void kernel_launch(void* const* d_in, const int* in_sizes, int n_in,
                              void* d_out, int out_size, void* d_ws, size_t ws_size,
                              hipStream_t stream) {
    const float* z_e = (const float*)d_in[0];   // (32, 64, 64, 64) f32
    const float* emb = (const float*)d_in[1];   // (512, 64) f32
    float* out = (float*)d_out;                 // z_q | indices | loss
    float* ws  = (float*)d_ws;
    float* norms    = ws;          // 512 floats
    float* partials = ws + 512;    // 512 floats

    vq_norms<<<2, 256, 0, stream>>>(emb, norms);

    const size_t smemBytes = (size_t)(32768 + 512 + 256) * sizeof(float); // ~131 KB
    vq_main<<<NBLOCKS, 256, smemBytes, stream>>>(z_e, emb, norms, out, partials);

    vq_finalize<<<1, 256, 0, stream>>>(partials, out);
}